// SimVQ1D_15161234555465
// MI455X (gfx1250) — compile-verified
//
#include <hip/hip_runtime.h>
#include <hip/hip_bf16.h>
#include <math.h>

// SimVQ forward for B=16,T=1024,C=32,K=16384 on gfx1250.
// Dominant cost: exact fp32 NN search (16384x16384x32 matmul) done with
// V_WMMA_F32_16X16X4_F32. Inputs (~4MB) are L2-resident; each block now
// covers 32 tokens (2 M-tiles per wave) so one streamed B tile feeds two
// independent WMMA accumulation chains -> half the L2 traffic, better
// hazard overlap between chained WMMAs.

typedef __attribute__((ext_vector_type(2))) float v2f;
typedef __attribute__((ext_vector_type(8))) float v8f;

static constexpr int kBT = 16384;   // B*T tokens
static constexpr int kC  = 32;      // code dim
static constexpr int kK  = 16384;   // codebook size

// ---------------- kernel 1: zero scratch ----------------
__global__ void simvq_zero(unsigned int* __restrict__ counts,
                           float* __restrict__ sumsq) {
    int i = blockIdx.x * blockDim.x + threadIdx.x;
    if (i < kK) counts[i] = 0u;
    if (i == 0) sumsq[0] = 0.0f;
}

// ---------------- kernel 2: cb = codebook @ W + bias; e_norm ----------------
// One wave32 per codebook row; lane j computes output column j.
__global__ __launch_bounds__(256) void simvq_project(
    const float* __restrict__ codebook, const float* __restrict__ W,
    const float* __restrict__ bias, float* __restrict__ cb,
    float* __restrict__ enorm) {
    const int lane = threadIdx.x & 31;
    const int wave = threadIdx.x >> 5;
    const int k    = blockIdx.x * 8 + wave;     // row of codebook
    float acc = bias[lane];
    #pragma unroll
    for (int c = 0; c < kC; ++c)
        acc += codebook[(size_t)k * kC + c] * W[(size_t)c * kC + lane];
    cb[(size_t)k * kC + lane] = acc;
    float sq = acc * acc;
    #pragma unroll
    for (int off = 16; off >= 1; off >>= 1)
        sq += __shfl_xor(sq, off, 32);
    if (lane == 0) enorm[k] = sq;
}

// ---------------- kernel 3: WMMA NN search + gather + partial loss ----------
// grid.x = BT/32 (two 16-row token tiles per block), 256 threads = 8 waves.
// score(m,n) = e_norm[n] - 2*dot(z_m, cb_n)   (z_norm constant per row).
__global__ __launch_bounds__(256) void simvq_search(
    const float* __restrict__ z, const float* __restrict__ cb,
    const float* __restrict__ enorm, float* __restrict__ out_zq,
    float* __restrict__ out_idx, unsigned int* __restrict__ counts,
    float* __restrict__ sumsq) {
    __shared__ float sd[8][32];
    __shared__ int   si[8][32];
    __shared__ int   srow[32];

    const int lane  = threadIdx.x & 31;
    const int wave  = threadIdx.x >> 5;
    const int m0    = blockIdx.x * 32;
    const int row16 = lane & 15;          // row/col within 16-lane half
    const int khalf = (lane >> 4) * 2;    // K sub-offset per ISA A/B layout

    // Preload both A tiles (2 x 16 tokens x 32), all 8 k-chunks each:
    // lane L holds z[mbase + (L&15)][4c + 2*(L>>4) + {0,1}] as one b64 load.
    v2f a0[8], a1[8];
    #pragma unroll
    for (int c = 0; c < 8; ++c) {
        a0[c] = *(const v2f*)(z + (size_t)(m0      + row16) * kC + 4 * c + khalf);
        a1[c] = *(const v2f*)(z + (size_t)(m0 + 16 + row16) * kC + 4 * c + khalf);
    }

    float best_d0[8], best_d1[8];
    int   best_i0[8], best_i1[8];
    #pragma unroll
    for (int r = 0; r < 8; ++r) {
        best_d0[r] = 3.4e38f; best_i0[r] = 0;
        best_d1[r] = 3.4e38f; best_i1[r] = 0;
    }

    // Each wave sweeps N-tiles wave, wave+8, ... (uniform trip count -> EXEC
    // stays all-ones across every WMMA).
    for (int nt = wave; nt < kK / 16; nt += 8) {
        const int n = nt * 16 + row16;    // this lane's code column
        v2f b[8];
        #pragma unroll
        for (int c = 0; c < 8; ++c)       // B = cb^T tile: contiguous in cb row
            b[c] = *(const v2f*)(cb + (size_t)n * kC + 4 * c + khalf);
        v8f acc0 = {0.f, 0.f, 0.f, 0.f, 0.f, 0.f, 0.f, 0.f};
        v8f acc1 = {0.f, 0.f, 0.f, 0.f, 0.f, 0.f, 0.f, 0.f};
        #pragma unroll
        for (int c = 0; c < 8; ++c) {     // two independent chains share b[c]
            acc0 = __builtin_amdgcn_wmma_f32_16x16x4_f32(
                false, a0[c], false, b[c], (short)0, acc0, false, false);
            acc1 = __builtin_amdgcn_wmma_f32_16x16x4_f32(
                false, a1[c], false, b[c], (short)0, acc1, false, false);
        }
        const float e = enorm[n];
        // Per-lane n is strictly increasing -> strict '<' keeps first min,
        // no tie-break compare needed inside the hot loop.
        #pragma unroll
        for (int r = 0; r < 8; ++r) {     // C layout: VGPR r = rows r / r+8
            const float d0 = fmaf(-2.0f, acc0[r], e);
            if (d0 < best_d0[r]) { best_d0[r] = d0; best_i0[r] = n; }
            const float d1 = fmaf(-2.0f, acc1[r], e);
            if (d1 < best_d1[r]) { best_d1[r] = d1; best_i1[r] = n; }
        }
    }

    // Argmin across the 16 lanes of each half (index tiebreak = first min).
    #pragma unroll
    for (int r = 0; r < 8; ++r) {
        #pragma unroll
        for (int off = 8; off >= 1; off >>= 1) {
            float od0 = __shfl_xor(best_d0[r], off, 32);
            int   oi0 = __shfl_xor(best_i0[r], off, 32);
            if (od0 < best_d0[r] || (od0 == best_d0[r] && oi0 < best_i0[r])) {
                best_d0[r] = od0; best_i0[r] = oi0;
            }
            float od1 = __shfl_xor(best_d1[r], off, 32);
            int   oi1 = __shfl_xor(best_i1[r], off, 32);
            if (od1 < best_d1[r] || (od1 == best_d1[r] && oi1 < best_i1[r])) {
                best_d1[r] = od1; best_i1[r] = oi1;
            }
        }
    }
    if ((lane & 15) == 0) {
        const int half = lane >> 4;       // lanes 0 / 16 hold rows r / r+8
        #pragma unroll
        for (int r = 0; r < 8; ++r) {
            sd[wave][r + 8 * half]      = best_d0[r];
            si[wave][r + 8 * half]      = best_i0[r];
            sd[wave][16 + r + 8 * half] = best_d1[r];
            si[wave][16 + r + 8 * half] = best_i1[r];
        }
    }
    __syncthreads();

    // Cross-wave argmin; emit index, bincount.
    if (threadIdx.x < 32) {
        float bd = sd[0][threadIdx.x];
        int   bi = si[0][threadIdx.x];
        #pragma unroll
        for (int w = 1; w < 8; ++w) {
            float od = sd[w][threadIdx.x];
            int   oi = si[w][threadIdx.x];
            if (od < bd || (od == bd && oi < bi)) { bd = od; bi = oi; }
        }
        srow[threadIdx.x] = bi;
        out_idx[m0 + threadIdx.x] = (float)bi;
        atomicAdd(&counts[bi], 1u);
    }
    __syncthreads();

    // Gather z_q = cb[idx] and accumulate sum((q-z)^2) for the loss.
    float part = 0.0f;
    #pragma unroll
    for (int e4 = threadIdx.x; e4 < 32 * kC; e4 += 256) {
        const int r = e4 >> 5, col = e4 & 31;
        const float q  = cb[(size_t)srow[r] * kC + col];
        const float zv = z[(size_t)(m0 + r) * kC + col];
        out_zq[(size_t)(m0 + r) * kC + col] = q;
        const float diff = q - zv;
        part += diff * diff;
    }
    #pragma unroll
    for (int off = 16; off >= 1; off >>= 1)
        part += __shfl_xor(part, off, 32);
    if (lane == 0) atomicAdd(sumsq, part);
}

// ---------------- kernel 4: stats finalize (single block) -------------------
__global__ __launch_bounds__(256) void simvq_finalize(
    const unsigned int* __restrict__ counts_ws,
    const float* __restrict__ sumsq, float* __restrict__ out) {
    float* o_commit = out + 524288;
    float* o_perp   = out + 524289;
    float* o_avg    = out + 524290;
    float* o_usage  = out + 540674;
    float* o_counts = out + 540675;
    float* o_total  = out + 557059;

    __shared__ float red[256];
    const int t = threadIdx.x;

    float tot = 0.0f, used = 0.0f;
    for (int k = t; k < kK; k += 256) {
        const float c = (float)counts_ws[k];
        o_counts[k] = c;
        tot += c;
        used += (c > 0.0f) ? 1.0f : 0.0f;
    }
    red[t] = tot; __syncthreads();
    for (int s = 128; s > 0; s >>= 1) { if (t < s) red[t] += red[t + s]; __syncthreads(); }
    const float total = fmaxf(red[0], 1.0f);
    __syncthreads();
    red[t] = used; __syncthreads();
    for (int s = 128; s > 0; s >>= 1) { if (t < s) red[t] += red[t + s]; __syncthreads(); }
    const float usedsum = red[0];
    __syncthreads();

    float h = 0.0f;  // sum p*log(p+eps) over nonzero p
    for (int k = t; k < kK; k += 256) {
        const float p = (float)counts_ws[k] / total;
        o_avg[k] = p;
        if (p > 0.0f) h += p * logf(p + 1e-10f);
    }
    red[t] = h; __syncthreads();
    for (int s = 128; s > 0; s >>= 1) { if (t < s) red[t] += red[t + s]; __syncthreads(); }
    if (t == 0) {
        o_perp[0]   = expf(-red[0]);
        o_usage[0]  = usedsum / (float)kK;
        o_total[0]  = total;
        // commit_loss = (BETA + 1) * mean((q - z)^2)
        o_commit[0] = 1.25f * sumsq[0] / (float)(kBT * kC);
    }
}

extern "C" void kernel_launch(void* const* d_in, const int* in_sizes, int n_in,
                              void* d_out, int out_size, void* d_ws, size_t ws_size,
                              hipStream_t stream) {
    const float* z        = (const float*)d_in[0];   // [16,1024,32]
    const float* codebook = (const float*)d_in[1];   // [16384,32]
    const float* W        = (const float*)d_in[2];   // [32,32]
    const float* bias     = (const float*)d_in[3];   // [32]
    float* out = (float*)d_out;

    // Workspace layout (floats): cb[K*C] | enorm[K] | counts[K] (u32) | sumsq[1]
    float* ws = (float*)d_ws;
    float*        cb     = ws;
    float*        enorm  = ws + (size_t)kK * kC;
    unsigned int* counts = (unsigned int*)(ws + (size_t)kK * kC + kK);
    float*        sumsq  = ws + (size_t)kK * kC + 2 * (size_t)kK;

    float* out_zq  = out;                 // [BT, C]
    float* out_idx = out + 557060;        // [BT] as float

    simvq_zero<<<64, 256, 0, stream>>>(counts, sumsq);
    simvq_project<<<kK / 8, 256, 0, stream>>>(codebook, W, bias, cb, enorm);
    simvq_search<<<kBT / 32, 256, 0, stream>>>(z, cb, enorm, out_zq, out_idx,
                                               counts, sumsq);
    simvq_finalize<<<1, 256, 0, stream>>>(counts, sumsq, out);
}